// Fastfood_Stack_Object_21251498180828
// MI455X (gfx1250) — compile-verified
//
#include <hip/hip_runtime.h>

typedef float v2f __attribute__((ext_vector_type(2)));
typedef float v4f __attribute__((ext_vector_type(4)));
typedef float v8f __attribute__((ext_vector_type(8)));

#define D      1024
#define WAVES  4          // 4 waves/block, one row per wave

// H16 entry: (-1)^popcount(r&c)
__device__ __forceinline__ float h16sign(int r, int c) {
    return (__builtin_popcount(r & c) & 1) ? -1.0f : 1.0f;
}

// dst[16g+i] = sum_k H16[i][k] * src[16g+k]   (H16 on bits 3..0)
__device__ __forceinline__ void wht_low(const float* __restrict__ src,
                                        float* __restrict__ dst,
                                        int m, int hi, int kbase, const v2f* af) {
    #pragma unroll
    for (int t = 0; t < 4; ++t) {
        const int g = 16 * t + m;
        v8f c = {0.f,0.f,0.f,0.f,0.f,0.f,0.f,0.f};
        #pragma unroll
        for (int kc = 0; kc < 4; ++kc) {
            const int k0 = 4 * kc + kbase;
            v2f b;
            b[0] = src[16 * g + k0];
            b[1] = src[16 * g + k0 + 1];
            c = __builtin_amdgcn_wmma_f32_16x16x4_f32(false, af[kc], false, b,
                                                      (short)0, c, false, false);
        }
        #pragma unroll
        for (int v = 0; v < 8; ++v)
            dst[16 * g + v + hi * 8] = c[v];
    }
}

// Same as wht_low but B fragments gather src[P[idx]] * G[idx] on the fly
// (fuses the permutation + G stage of the Fastfood pipeline).
__device__ __forceinline__ void wht_low_perm(const float* __restrict__ src,
                                             float* __restrict__ dst,
                                             const int* __restrict__ Pl,
                                             const float* __restrict__ Gl,
                                             int m, int hi, int kbase, const v2f* af) {
    #pragma unroll
    for (int t = 0; t < 4; ++t) {
        const int g = 16 * t + m;
        v8f c = {0.f,0.f,0.f,0.f,0.f,0.f,0.f,0.f};
        #pragma unroll
        for (int kc = 0; kc < 4; ++kc) {
            const int k0 = 4 * kc + kbase;
            const int i0 = 16 * g + k0;
            v2f b;
            b[0] = src[Pl[i0]]     * Gl[i0];
            b[1] = src[Pl[i0 + 1]] * Gl[i0 + 1];
            c = __builtin_amdgcn_wmma_f32_16x16x4_f32(false, af[kc], false, b,
                                                      (short)0, c, false, false);
        }
        #pragma unroll
        for (int v = 0; v < 8; ++v)
            dst[16 * g + v + hi * 8] = c[v];
    }
}

// dst[64i + 16tt + m] : H16 on bits 9..6 via WMMA, then H4 on bits 5..4
// fused as an intra-lane radix-4 combine of the four tile accumulators
// (tile index t == output bits 5..4, held in the same lane for fixed m).
__device__ __forceinline__ void wht_high_mid(const float* __restrict__ src,
                                             float* __restrict__ dst,
                                             int m, int hi, int kbase, const v2f* af) {
    v8f c0 = {0.f,0.f,0.f,0.f,0.f,0.f,0.f,0.f};
    v8f c1 = c0, c2 = c0, c3 = c0;
    #pragma unroll
    for (int kc = 0; kc < 4; ++kc) {
        const int k0 = 4 * kc + kbase;
        const float* s0 = src + 64 * k0;
        const float* s1 = src + 64 * (k0 + 1);
        v2f b;
        b[0] = s0[m];       b[1] = s1[m];
        c0 = __builtin_amdgcn_wmma_f32_16x16x4_f32(false, af[kc], false, b, (short)0, c0, false, false);
        b[0] = s0[m + 16];  b[1] = s1[m + 16];
        c1 = __builtin_amdgcn_wmma_f32_16x16x4_f32(false, af[kc], false, b, (short)0, c1, false, false);
        b[0] = s0[m + 32];  b[1] = s1[m + 32];
        c2 = __builtin_amdgcn_wmma_f32_16x16x4_f32(false, af[kc], false, b, (short)0, c2, false, false);
        b[0] = s0[m + 48];  b[1] = s1[m + 48];
        c3 = __builtin_amdgcn_wmma_f32_16x16x4_f32(false, af[kc], false, b, (short)0, c3, false, false);
    }
    // H4 across tiles, purely in registers (Sylvester rows ++++, +-+-, ++--, +--+)
    const v8f d0 = c0 + c1 + c2 + c3;
    const v8f d1 = c0 - c1 + c2 - c3;
    const v8f d2 = c0 + c1 - c2 - c3;
    const v8f d3 = c0 - c1 - c2 + c3;
    #pragma unroll
    for (int v = 0; v < 8; ++v) {
        float* p = dst + 64 * (v + hi * 8) + m;
        p[0]  = d0[v];
        p[16] = d1[v];
        p[32] = d2[v];
        p[48] = d3[v];
    }
}

__global__ __launch_bounds__(WAVES * 32)
void fastfood_kernel(const float* __restrict__ x, const float* __restrict__ B,
                     const float* __restrict__ G, const float* __restrict__ S,
                     const int* __restrict__ P, float* __restrict__ out, int rows) {
    __shared__ float buf[2][WAVES][D];     // ping-pong per wave
    __shared__ int   Pl[D];                // block-shared permutation
    __shared__ float Gl[D];                // block-shared G

    const int tid  = threadIdx.x;
    const int lane = tid & 31;
    const int wv   = tid >> 5;
    const int row  = blockIdx.x * WAVES + wv;
    const bool active = (row < rows);      // wave-uniform -> EXEC all-1s at WMMAs
    const int m     = lane & 15;
    const int hi    = lane >> 4;
    const int kbase = hi * 2;

    // stage P/G once per block (their use pattern below is scattered)
    for (int i = tid; i < D; i += WAVES * 32) { Pl[i] = P[i]; Gl[i] = G[i]; }

    // A fragments (H16 column chunks) — invariant across all 64 WMMAs
    v2f af[4];
    #pragma unroll
    for (int kc = 0; kc < 4; ++kc) {
        const int k0 = 4 * kc + kbase;
        af[kc][0] = h16sign(m, k0);
        af[kc][1] = h16sign(m, k0 + 1);
    }

    float* b0 = buf[0][wv];
    float* b1 = buf[1][wv];

    // ---- load row (streamed once, non-temporal) * B -> b0
    if (active) {
        const v4f* xr = (const v4f*)(x + (size_t)row * D);
        const v4f* B4 = (const v4f*)B;
        v4f* d4 = (v4f*)b0;
        for (int i = lane; i < D / 4; i += 32)
            d4[i] = __builtin_nontemporal_load(&xr[i]) * B4[i];
    }
    __syncthreads();

    // ---- WHT #1
    if (active) wht_low(b0, b1, m, hi, kbase, af);
    __syncthreads();
    if (active) wht_high_mid(b1, b0, m, hi, kbase, af);          // b0 = H(Bx)
    __syncthreads();

    // ---- WHT #2 with fused permutation + G on the input gather
    if (active) wht_low_perm(b0, b1, Pl, Gl, m, hi, kbase, af);
    __syncthreads();
    if (active) wht_high_mid(b1, b0, m, hi, kbase, af);          // b0 = H(G·perm)
    __syncthreads();

    // ---- scale by S / sqrt(d), non-temporal store
    if (active) {
        v4f* o4 = (v4f*)(out + (size_t)row * D);
        const v4f* S4 = (const v4f*)S;
        const v4f* s0 = (const v4f*)b0;
        for (int i = lane; i < D / 4; i += 32)
            __builtin_nontemporal_store(s0[i] * S4[i] * 0.03125f, &o4[i]);
    }
}

extern "C" void kernel_launch(void* const* d_in, const int* in_sizes, int n_in,
                              void* d_out, int out_size, void* d_ws, size_t ws_size,
                              hipStream_t stream) {
    const float* x = (const float*)d_in[0];
    const float* B = (const float*)d_in[1];
    const float* G = (const float*)d_in[2];
    const float* S = (const float*)d_in[3];
    const int*   P = (const int*)  d_in[4];
    float* out = (float*)d_out;

    const int rows = in_sizes[0] / D;                  // 65536 for the given shape
    const int grid = (rows + WAVES - 1) / WAVES;
    fastfood_kernel<<<grid, WAVES * 32, 0, stream>>>(x, B, G, S, P, out, rows);
}